// DynamicRelationModeler_71382356459939
// MI455X (gfx1250) — compile-verified
//
#include <hip/hip_runtime.h>
#include <math.h>

typedef __attribute__((ext_vector_type(2))) float v2f;
typedef __attribute__((ext_vector_type(8))) float v8f;

#define NN   512
#define FF   512
#define HH   512
#define RNUM 4
#define NH   (512*512)

// ---------------- Kernel 0: x = concat(feats_img, feats_txt) -> d_out[0:NH)
__global__ void concat_kernel(const float* __restrict__ img,
                              const float* __restrict__ txt,
                              float* __restrict__ x) {
    int idx = blockIdx.x * blockDim.x + threadIdx.x;   // 0..262143
    x[idx] = (idx < 256 * FF) ? img[idx] : txt[idx - 256 * FF];
}

// ---------------- Kernel 1: 8 fp32 GEMMs via v_wmma_f32_16x16x4_f32
// ws[r]       = x @ Wl[r]^T            (side 0)
// ws[4+r]     = x @ Wr[r]^T + b1[r]    (side 1)
__global__ __launch_bounds__(128)
void gemm_wmma_f32(const float* __restrict__ x,
                   const float* __restrict__ W1,   // [R, H, 2F]
                   const float* __restrict__ b1,   // [R, H]
                   float* __restrict__ ws) {
    const int wave = (blockIdx.x << 2) | (threadIdx.x >> 5); // 0..8191
    const int lane = threadIdx.x & 31;
    const int mat  = wave >> 10;          // 0..7 -> (r, side)
    const int rem  = wave & 1023;
    const int mt   = rem >> 5;            // 0..31 row-tile
    const int ht   = rem & 31;            // 0..31 col-tile
    const int r    = mat >> 1;
    const int side = mat & 1;
    const int m0   = mt << 4;
    const int h0   = ht << 4;
    const int lo   = lane & 15;
    const int hi   = lane >> 4;

    // A (16x4 f32): lane holds A[m][2*hi + v] -> contiguous float2 along K
    const float* arow = x + (m0 + lo) * FF + 2 * hi;
    // B (4x16 f32): lane holds B[2*hi + v][n] = W[h0+n][kk + 2*hi + v]
    const float* brow = W1 + ((size_t)(r * HH + h0 + lo)) * (2 * FF) + side * FF + 2 * hi;

    v8f acc = {};
    for (int kk = 0; kk < FF; kk += 4) {
        v2f a = *(const v2f*)(arow + kk);
        v2f b = *(const v2f*)(brow + kk);
        acc = __builtin_amdgcn_wmma_f32_16x16x4_f32(
            /*neg_a=*/false, a, /*neg_b=*/false, b,
            /*c_mod=*/(short)0, acc, /*reuse_a=*/false, /*reuse_b=*/false);
    }

    const float bias = side ? b1[r * HH + h0 + lo] : 0.0f;
    float* out = ws + (size_t)(side * 4 + r) * NH;
    // C/D layout: VGPR v -> M = v + 8*hi, N = lane&15
    #pragma unroll
    for (int v = 0; v < 8; ++v) {
        int row = m0 + v + (hi << 3);
        out[(size_t)row * HH + h0 + lo] = acc[v] + bias;
    }
}

// ---------------- Kernel 2: pairwise LN + ReLU + dot, argmax over r
__global__ __launch_bounds__(256)
void pairwise_kernel(const float* __restrict__ ws,
                     const float* __restrict__ gamma,  // [R,H]
                     const float* __restrict__ beta,   // [R,H]
                     const float* __restrict__ w2,     // [R,H]
                     const float* __restrict__ b2,     // [R]
                     float* __restrict__ out) {
    __shared__ __align__(16) float a_sh[16 * HH];   // 32 KB
    __shared__ __align__(16) float c_sh[16 * HH];   // 32 KB
    __shared__ __align__(16) float g_sh[HH];
    __shared__ __align__(16) float be_sh[HH];
    __shared__ __align__(16) float wv_sh[HH];

    const int tid = threadIdx.x;
    const int jt = blockIdx.x, it = blockIdx.y;
    const int i0 = it << 4, j0 = jt << 4;
    const int ti = tid >> 4, tj = tid & 15;
    const int i = i0 + ti, j = j0 + tj;

    float best = -INFINITY;
    int   bestr = 0;

    for (int r = 0; r < RNUM; ++r) {
        __syncthreads();  // protect LDS from previous iteration's readers
        const float* Ar = ws + (size_t)r * NH;
        const float* Cr = ws + (size_t)(4 + r) * NH;
        // 16 rows x 128 float4 per tile, 256 threads -> 8 float4 each
        for (int idx = tid; idx < 16 * (HH / 4); idx += 256) {
            int row = idx >> 7, c4 = idx & 127;
            ((float4*)a_sh)[row * (HH / 4) + c4] =
                ((const float4*)(Ar + (size_t)(i0 + row) * HH))[c4];
            ((float4*)c_sh)[row * (HH / 4) + c4] =
                ((const float4*)(Cr + (size_t)(j0 + row) * HH))[c4];
        }
        for (int idx = tid; idx < HH; idx += 256) {
            g_sh[idx]  = gamma[r * HH + idx];
            be_sh[idx] = beta[r * HH + idx];
            wv_sh[idx] = w2[r * HH + idx];
        }
        __syncthreads();

        const float4* av = (const float4*)(a_sh + ti * HH);
        const float4* cv = (const float4*)(c_sh + tj * HH);

        // pass 1: mean / variance of h = a_i + c_j  (b1 already folded into c)
        float s1 = 0.0f, s2 = 0.0f;
        for (int k = 0; k < HH / 4; ++k) {
            float4 a = av[k], c = cv[k];
            float h0 = a.x + c.x, h1 = a.y + c.y, h2 = a.z + c.z, h3 = a.w + c.w;
            s1 += (h0 + h1) + (h2 + h3);
            s2 += (h0 * h0 + h1 * h1) + (h2 * h2 + h3 * h3);
        }
        float mu  = s1 * (1.0f / HH);
        float var = s2 * (1.0f / HH) - mu * mu;
        float rs  = rsqrtf(var + 1e-5f);

        // pass 2: relu(LN(h)) . w2
        float acc = 0.0f;
        for (int k = 0; k < HH / 4; ++k) {
            float4 a = av[k], c = cv[k];
            float4 g = ((const float4*)g_sh)[k];
            float4 be = ((const float4*)be_sh)[k];
            float4 wv = ((const float4*)wv_sh)[k];
            float h0 = (a.x + c.x - mu) * rs * g.x + be.x;
            float h1 = (a.y + c.y - mu) * rs * g.y + be.y;
            float h2 = (a.z + c.z - mu) * rs * g.z + be.z;
            float h3 = (a.w + c.w - mu) * rs * g.w + be.w;
            acc += fmaxf(h0, 0.0f) * wv.x + fmaxf(h1, 0.0f) * wv.y
                 + fmaxf(h2, 0.0f) * wv.z + fmaxf(h3, 0.0f) * wv.w;
        }
        float s = acc + b2[r];
        if (s > best) { best = s; bestr = r; }   // strict > keeps first max (jnp.argmax)
    }

    const size_t p = (size_t)i * NN + j;
    out[NH + p]     = best;
    out[2 * NH + p] = (float)bestr;
    out[3 * NH + p] = (i != j && best > 0.2f) ? 1.0f : 0.0f;
}

extern "C" void kernel_launch(void* const* d_in, const int* in_sizes, int n_in,
                              void* d_out, int out_size, void* d_ws, size_t ws_size,
                              hipStream_t stream) {
    const float* img   = (const float*)d_in[0];
    const float* txt   = (const float*)d_in[1];
    const float* W1    = (const float*)d_in[2];
    const float* b1    = (const float*)d_in[3];
    const float* gamma = (const float*)d_in[4];
    const float* beta  = (const float*)d_in[5];
    const float* w2    = (const float*)d_in[6];
    const float* b2    = (const float*)d_in[7];
    float* out = (float*)d_out;
    float* ws  = (float*)d_ws;   // needs 8 * 512 * 512 * 4 B = 8 MB

    // 1) x = concat(img, txt) into d_out[0:NH)
    concat_kernel<<<NH / 256, 256, 0, stream>>>(img, txt, out);
    // 2) 8 fp32 WMMA GEMMs: A_r and C_r(+b1) into workspace
    gemm_wmma_f32<<<2048, 128, 0, stream>>>(out, W1, b1, ws);
    // 3) pairwise LN/ReLU/dot + argmax + threshold mask
    dim3 grid(NN / 16, NN / 16);
    pairwise_kernel<<<grid, 256, 0, stream>>>(ws, gamma, beta, w2, b2, out);
}